// Prototypes_27152783245865
// MI455X (gfx1250) — compile-verified
//
#include <hip/hip_runtime.h>

// Prototype L2-distance min/argmin over patches, via WMMA f32 GEMM expansion:
//   dist^2(m,n) = ||x_m||^2 - 2 x_m . p_n + ||p_n||^2
// x: [32, 512, 64] f32, prototypes: [256, 64] f32
// out: dist [32,1,256] f32  ++  argmin [32,1,256] int32

typedef __attribute__((ext_vector_type(2))) float v2f;
typedef __attribute__((ext_vector_type(8))) float v8f;

#define PROTO_EPS 1e-8f

__global__ __launch_bounds__(128) void Prototypes_27152783245865_kernel(
    const float* __restrict__ x, const float* __restrict__ protos,
    float* __restrict__ out_dist, int* __restrict__ out_idx)
{
    const int lane  = threadIdx.x & 31;        // lane in wave32
    const int wave  = threadIdx.x >> 5;        // 0..3
    const int ktile = (int)blockIdx.x * 4 + wave;  // 0..15 (16 proto tiles)
    const int b     = (int)blockIdx.y;             // 0..31 batches
    const int ln    = lane & 15;               // row within tile
    const int h     = lane >> 4;               // half-wave: selects K pair / M half

    // ---- Load B fragments: 16 prototypes x 64 dims, f32 WMMA 16x16x4 layout.
    // Lane holds proto row (ln), float2 at dim offset 4*i + 2*h for WMMA step i.
    const float* prow = protos + (size_t)(ktile * 16 + ln) * 64 + 2 * h;
    v2f pb[16];
    float psq_part = 0.0f;
#pragma unroll
    for (int i = 0; i < 16; ++i) {
        pb[i] = *(const v2f*)(prow + 4 * i);
        psq_part += pb[i].x * pb[i].x + pb[i].y * pb[i].y;
    }
    // Each lane has half of ||p_n||^2 for n = ln; other half sits in lane^16.
    float psq = psq_part + __shfl_xor(psq_part, 16, 32);

    float best = 3.402823466e38f;
    int   bidx = 0;

    // ---- Sweep 32 patch tiles of 16 patches each (512 patches total).
    for (int pt = 0; pt < 32; ++pt) {
        const float* xrow =
            x + ((size_t)(b * 512 + pt * 16 + ln) * 64) + 2 * h;
        v2f xa[16];
        float xsq_part = 0.0f;
#pragma unroll
        for (int i = 0; i < 16; ++i) {
            xa[i] = *(const v2f*)(xrow + 4 * i);
            xsq_part += xa[i].x * xa[i].x + xa[i].y * xa[i].y;
        }
        float xsqf = xsq_part + __shfl_xor(xsq_part, 16, 32);  // full ||x_m||^2, m = ln

        // ---- D = X(16x64) * P^T(64x16) via 16 chained f32 16x16x4 WMMAs.
        v8f acc = {};
#pragma unroll
        for (int i = 0; i < 16; ++i) {
            acc = __builtin_amdgcn_wmma_f32_16x16x4_f32(
                /*neg_a=*/false, xa[i], /*neg_b=*/false, pb[i],
                /*c_mod=*/(short)0, acc, /*reuse_a=*/false, /*reuse_b=*/false);
        }

        // ---- Fold into running min/argmin. Lane covers M = v + 8*h, N = ln.
#pragma unroll
        for (int v = 0; v < 8; ++v) {
            int   m    = v + 8 * h;
            float xs   = __shfl(xsqf, m, 32);          // ||x_m||^2 from lane m
            float d2   = xs - 2.0f * acc[v] + psq;
            float dist = __builtin_sqrtf(fmaxf(d2, 0.0f) + PROTO_EPS);
            int   gm   = pt * 16 + m;
            if (dist < best) { best = dist; bidx = gm; }  // strict <: first occurrence
        }
    }

    // ---- Merge the two M-halves (lane L vs L+16 share the same N).
    float ob = __shfl_xor(best, 16, 32);
    int   oi = __shfl_xor(bidx, 16, 32);
    if (ob < best || (ob == best && oi < bidx)) { best = ob; bidx = oi; }

    if (lane < 16) {
        int k = ktile * 16 + ln;
        out_dist[b * 256 + k] = best;
        out_idx [b * 256 + k] = bidx;
    }
}

extern "C" void kernel_launch(void* const* d_in, const int* in_sizes, int n_in,
                              void* d_out, int out_size, void* d_ws, size_t ws_size,
                              hipStream_t stream) {
    const float* x      = (const float*)d_in[0];   // [32, 512, 64]
    const float* protos = (const float*)d_in[1];   // [256, 64]
    float* out_dist = (float*)d_out;               // [32*256] f32
    int*   out_idx  = (int*)d_out + 32 * 256;      // [32*256] int32 (bit-cast region)

    dim3 grid(4, 32);   // 4 ktile-groups x 32 batches; 4 waves/block -> 16 ktiles
    Prototypes_27152783245865_kernel<<<grid, 128, 0, stream>>>(x, protos, out_dist, out_idx);
}